// KNNGrouper_7962869367538
// MI455X (gfx1250) — compile-verified
//
#include <hip/hip_runtime.h>
#include <stdint.h>

// ---------------------------------------------------------------------------
// Problem constants (match reference)
// ---------------------------------------------------------------------------
#define B_   4
#define N_   32768
#define C_   384
#define G_   1024
#define K_   32
#define OUTC (3 + 2 * C_)          // 771
#define INV_RADIUS 20.0f           // 1 / 0.05

// d_out layout (flat, return order): group_feats | centers | knn_idx | fps_idx
#define OFF_CENT ((size_t)B_ * G_ * K_ * OUTC)          // 101,056,512
#define OFF_KNN  (OFF_CENT + (size_t)B_ * G_ * 3)       // +12,288
#define OFF_FPS  (OFF_KNN + (size_t)B_ * G_ * K_)       // +131,072

typedef __attribute__((ext_vector_type(2))) float v2f;
typedef __attribute__((ext_vector_type(8))) float v8f;

// ---------------------------------------------------------------------------
// Kernel 1: farthest point sampling. One 1024-thread block per batch.
// Each thread owns 32 points; running min-distance lives in registers.
// Per step: distance update + block argmax (wave shuffle + LDS tree).
// ---------------------------------------------------------------------------
__global__ __launch_bounds__(1024) void fps_kernel(const float* __restrict__ xyz,
                                                   float* __restrict__ cent,
                                                   int* __restrict__ fpsi) {
    const int b   = blockIdx.x;
    const int tid = threadIdx.x;
    const float* X = xyz + (size_t)b * N_ * 3;

    __shared__ float s_last[3];
    __shared__ float s_wd[32];
    __shared__ int   s_wi[32];

    constexpr int PTS = N_ / 1024;   // 32 points per thread
    float dist[PTS];
#pragma unroll
    for (int t = 0; t < PTS; ++t) dist[t] = 1e10f;

    if (tid == 0) {
        float x0 = X[0], y0 = X[1], z0 = X[2];
        s_last[0] = x0; s_last[1] = y0; s_last[2] = z0;
        fpsi[b * G_ + 0] = 0;
        float* c0 = cent + (size_t)(b * G_) * 3;
        c0[0] = x0; c0[1] = y0; c0[2] = z0;
    }
    __syncthreads();

    const int lane = tid & 31;
    const int wid  = tid >> 5;

    for (int i = 1; i < G_; ++i) {
        const float lx = s_last[0], ly = s_last[1], lz = s_last[2];
        float bd = -1.0f;
        int   bp = 0;
#pragma unroll
        for (int t = 0; t < PTS; ++t) {
            const int p = t * 1024 + tid;
            const float* xp = X + (size_t)p * 3;
            const float dx = xp[0] - lx, dy = xp[1] - ly, dz = xp[2] - lz;
            const float d  = fmaf(dx, dx, fmaf(dy, dy, dz * dz));
            const float nd = fminf(dist[t], d);
            dist[t] = nd;
            if (nd > bd) { bd = nd; bp = p; }  // strict > keeps lowest index on ties
        }
        // wave32 argmax reduce (tie -> lower index, matching jnp.argmax)
#pragma unroll
        for (int off = 16; off > 0; off >>= 1) {
            const float od = __shfl_xor(bd, off, 32);
            const int   op = __shfl_xor(bp, off, 32);
            if (od > bd || (od == bd && op < bp)) { bd = od; bp = op; }
        }
        if (lane == 0) { s_wd[wid] = bd; s_wi[wid] = bp; }
        __syncthreads();
        if (wid == 0) {
            bd = s_wd[lane]; bp = s_wi[lane];
#pragma unroll
            for (int off = 16; off > 0; off >>= 1) {
                const float od = __shfl_xor(bd, off, 32);
                const int   op = __shfl_xor(bp, off, 32);
                if (od > bd || (od == bd && op < bp)) { bd = od; bp = op; }
            }
            if (lane == 0) {
                const float* xp = X + (size_t)bp * 3;
                s_last[0] = xp[0]; s_last[1] = xp[1]; s_last[2] = xp[2];
                fpsi[b * G_ + i] = bp;
                float* cg = cent + (size_t)(b * G_ + i) * 3;
                cg[0] = xp[0]; cg[1] = xp[1]; cg[2] = xp[2];
            }
        }
        __syncthreads();
    }
}

// ---------------------------------------------------------------------------
// Kernel 2: KNN via WMMA f32 16x16x4 + async-to-LDS point staging.
// One wave owns a tile of 16 centers; the block cooperatively double-buffers
// 128-point xyz chunks into LDS with global_load_async_to_lds_b128 (prefetch
// of chunk c+1 overlaps compute on chunk c).  Per chunk of 16 points one
// v_wmma produces the 16x16 dot-product tile; per-row top-32 lives in LDS
// with a cached threshold (fast path: 8 fma + 8 max + 1 ballot).
// ---------------------------------------------------------------------------
#define TILES 4    // waves (tiles of 16 centers) per block
#define CH    128  // points per staged chunk (128*12B = 1536B = 96 x b128)

__global__ __launch_bounds__(32 * TILES) void knn_kernel(const float* __restrict__ xyz,
                                                         const float* __restrict__ cent,
                                                         int* __restrict__ knn) {
    __shared__ float s_topd[TILES][16][K_];
    __shared__ int   s_topi[TILES][16][K_];
    __shared__ float s_rmax[TILES][16];
    __shared__ __align__(16) float s_pts[2][CH * 3];   // double-buffered xyz chunks

    const int b    = blockIdx.y;
    const int tid  = threadIdx.x;
    const int t    = tid >> 5;               // wave = tile slot
    const int lane = tid & 31;
    const int half = lane >> 4;              // D rows: M = v + 8*half
    const int col  = lane & 15;
    const int g0   = (blockIdx.x * TILES + t) * 16;

    for (int i = tid; i < TILES * 16 * K_; i += 32 * TILES) {
        (&s_topd[0][0][0])[i] = 1e30f;
        (&s_topi[0][0][0])[i] = 0;
    }
    for (int i = tid; i < TILES * 16; i += 32 * TILES)
        (&s_rmax[0][0])[i] = 1e30f;

    const float* X = xyz + (size_t)b * N_ * 3;
    const uint64_t xbase = (uint64_t)(uintptr_t)X;
    // LDS byte offsets of the two staging buffers (generic addr low 32 bits)
    const unsigned lds0 = (unsigned)(uintptr_t)&s_pts[0][0];
    const unsigned lds1 = (unsigned)(uintptr_t)&s_pts[1][0];

    // async-stage chunk c (1536B) into buffer buf: 96 lanes x b128
    auto stage = [&](int c, int buf) {
        if (tid < (CH * 12) / 16) {          // 96 transfers
            const unsigned l  = (buf ? lds1 : lds0) + (unsigned)tid * 16u;
            const uint64_t ga = xbase + (uint64_t)c * (CH * 12) + (uint64_t)tid * 16u;
            asm volatile("global_load_async_to_lds_b128 %0, %1, off"
                         :: "v"(l), "v"(ga) : "memory");
        }
    };

    // A operand (16x4 f32): lanes 0-15 -> {K0=x, K1=y}; lanes 16-31 -> {K2=z, K3=0}
    const float* crow = cent + (size_t)(b * G_ + g0 + col) * 3;
    const float cax = crow[0], cay = crow[1], caz = crow[2];
    v2f A;
    A.x = half ? caz : cax;
    A.y = half ? 0.0f : cay;

    float cn2r[8];
#pragma unroll
    for (int v = 0; v < 8; ++v) {
        const float* cr = cent + (size_t)(b * G_ + g0 + v + 8 * half) * 3;
        cn2r[v] = fmaf(cr[0], cr[0], fmaf(cr[1], cr[1], cr[2] * cr[2]));
    }
    float rmax[8];
#pragma unroll
    for (int v = 0; v < 8; ++v) rmax[v] = 1e30f;

    stage(0, 0);
    asm volatile("s_wait_asynccnt 0x0" ::: "memory");
    __syncthreads();                         // buffer 0 ready, LDS lists zeroed

    constexpr int NCH = N_ / CH;             // 256 chunks
    for (int c = 0; c < NCH; ++c) {
        const int buf = c & 1;
        if (c + 1 < NCH) stage(c + 1, buf ^ 1);   // overlap prefetch with compute

        const float* sp0 = &s_pts[buf][0];
        for (int pp = 0; pp < CH; pp += 16) {
            const int   pl = pp + col;
            const float* sp = sp0 + pl * 3;       // LDS read
            const float px = sp[0], py = sp[1], pz = sp[2];
            const float px2 = fmaf(px, px, fmaf(py, py, pz * pz));
            const int   p  = c * CH + pl;         // global point index
            // B operand (4x16 f32): lanes 0-15 -> {K0=x,K1=y}; 16-31 -> {K2=z,K3=0}
            v2f Bm;
            Bm.x = half ? pz : px;
            Bm.y = half ? 0.0f : py;

            v8f acc = {0.f, 0.f, 0.f, 0.f, 0.f, 0.f, 0.f, 0.f};
            acc = __builtin_amdgcn_wmma_f32_16x16x4_f32(
                /*neg_a=*/false, A, /*neg_b=*/false, Bm,
                /*c_mod=*/(short)0, acc, /*reuse_a=*/false, /*reuse_b=*/false);

            float d2v[8];
            float cnd = -1.0f;                    // max over v of (rmax - d2)
#pragma unroll
            for (int v = 0; v < 8; ++v) {
                const float d2 = fmaf(-2.0f, acc[v], cn2r[v] + px2);
                d2v[v] = d2;
                cnd = fmaxf(cnd, rmax[v] - d2);
            }
            unsigned long long m = __ballot(cnd > 0.0f);
            if (m) {
                // rare slow path: serialized single-lane insertions into LDS lists
                while (m) {
                    const int j = __ffsll(m) - 1;
                    m &= m - 1;
                    if (lane == j) {
#pragma unroll
                        for (int v = 0; v < 8; ++v) {
                            const int r = v + 8 * half;
                            if (d2v[v] < s_rmax[t][r]) {
                                float* td = &s_topd[t][r][0];
                                int*   ti = &s_topi[t][r][0];
                                int am = 0; float mx = td[0];
                                for (int s = 1; s < K_; ++s) {
                                    const float vv = td[s];
                                    if (vv > mx) { mx = vv; am = s; }
                                }
                                if (d2v[v] < mx) {
                                    td[am] = d2v[v]; ti[am] = p;
                                    float nm = td[0];
                                    for (int s = 1; s < K_; ++s) nm = fmaxf(nm, td[s]);
                                    s_rmax[t][r] = nm;
                                }
                            }
                        }
                    }
                }
#pragma unroll
                for (int v = 0; v < 8; ++v) rmax[v] = s_rmax[t][v + 8 * half];
            }
        }
        // next buffer landed; all waves done reading current before it's reused
        asm volatile("s_wait_asynccnt 0x0" ::: "memory");
        __syncthreads();
    }

    // Sort each row's 32 survivors ascending by d2 (tie -> lower idx), emit.
    for (int r = 0; r < 16; ++r) {
        float d   = s_topd[t][r][lane];
        int   idx = s_topi[t][r][lane];
        int* out  = knn + ((size_t)(b * G_) + g0 + r) * K_;
        for (int k = 0; k < K_; ++k) {
            float bd2 = d; int bi = idx;
#pragma unroll
            for (int off = 16; off > 0; off >>= 1) {
                const float od = __shfl_xor(bd2, off, 32);
                const int   oi = __shfl_xor(bi, off, 32);
                if (od < bd2 || (od == bd2 && oi < bi)) { bd2 = od; bi = oi; }
            }
            if (lane == 0) out[k] = bi;
            if (d == bd2 && idx == bi) d = 3.0e38f;  // retire winner
        }
    }
}

// ---------------------------------------------------------------------------
// Kernel 3: gather + concat.  One 256-thread block per (b,g) group.
// Streams ~404MB of output with nontemporal stores; prefetches gathered rows.
// ---------------------------------------------------------------------------
__global__ __launch_bounds__(256) void gather_kernel(const float* __restrict__ xyz,
                                                     const float* __restrict__ feat,
                                                     const float* __restrict__ cent,
                                                     const int* __restrict__ knn,
                                                     const int* __restrict__ fpsi,
                                                     float* __restrict__ gf) {
    const int g = blockIdx.x;
    const int b = blockIdx.y;
    const int tid = threadIdx.x;

    __shared__ float s_cf[C_];
    __shared__ int   s_idx[K_];
    __shared__ float s_c[3];

    if (tid < K_) s_idx[tid] = knn[((size_t)(b * G_) + g) * K_ + tid];
    if (tid < 3)  s_c[tid]   = cent[((size_t)(b * G_) + g) * 3 + tid];
    {
        const int fi = fpsi[b * G_ + g];
        const float* cf = feat + ((size_t)b * N_ + fi) * (size_t)C_;
        for (int c = tid; c < C_; c += 256) s_cf[c] = cf[c];
    }
    __syncthreads();

    const int lane = tid & 31;
    const int wid  = tid >> 5;

    for (int k = wid; k < K_; k += 8) {
        const int nb = s_idx[k];
        const float* nf = feat + ((size_t)b * N_ + nb) * (size_t)C_;
        float* orow = gf + (((size_t)(b * G_) + g) * K_ + k) * (size_t)OUTC;

        if (lane < 3) {
            const float nx = xyz[((size_t)b * N_ + nb) * 3 + lane];
            __builtin_nontemporal_store((nx - s_c[lane]) * INV_RADIUS, &orow[lane]);
        }
        // prefetch the feature row we'll gather next time around this loop
        if (k + 8 < K_) {
            const float* pf = feat + ((size_t)b * N_ + s_idx[k + 8]) * (size_t)C_;
            if (lane < 12) __builtin_prefetch(pf + lane * 32, 0, 3);  // 12 x 128B
        }
#pragma unroll
        for (int c = lane; c < C_; c += 32) {     // 12 coalesced iterations
            const float f = nf[c];
            __builtin_nontemporal_store(f, &orow[3 + c]);
            __builtin_nontemporal_store(f - s_cf[c], &orow[3 + C_ + c]);
        }
    }
}

// ---------------------------------------------------------------------------
extern "C" void kernel_launch(void* const* d_in, const int* in_sizes, int n_in,
                              void* d_out, int out_size, void* d_ws, size_t ws_size,
                              hipStream_t stream) {
    const float* xyz  = (const float*)d_in[0];   // [B,N,3]
    const float* feat = (const float*)d_in[1];   // [B,N,C]

    float* out  = (float*)d_out;
    float* gf   = out;                           // [B,G,K,771]
    float* cent = out + OFF_CENT;                // [B,G,3]
    int*   knn  = (int*)(out + OFF_KNN);         // [B,G,K]
    int*   fpsi = (int*)(out + OFF_FPS);         // [B,G]

    fps_kernel<<<dim3(B_), dim3(1024), 0, stream>>>(xyz, cent, fpsi);
    knn_kernel<<<dim3(G_ / (16 * TILES), B_), dim3(32 * TILES), 0, stream>>>(xyz, cent, knn);
    gather_kernel<<<dim3(G_, B_), dim3(256), 0, stream>>>(xyz, feat, cent, knn, fpsi, gf);
}